// BilinearPairedLayer_83373905150128
// MI455X (gfx1250) — compile-verified
//
#include <hip/hip_runtime.h>
#include <cstdint>

#define NB   8
#define NN   1024
#define F_IN 64
#define HH   128
#define KOUT 2

#define ITILE 32
#define JTILE 32
#define LDP   132   // 128 + 4-dword pad -> bank-conflict-free row access

typedef float v2f __attribute__((ext_vector_type(2)));
typedef float v8f __attribute__((ext_vector_type(8)));
typedef int   v4i_vs __attribute__((vector_size(16)));   // matches builtin param type

#if defined(__has_builtin)
#  if __has_builtin(__builtin_amdgcn_global_load_async_to_lds_b128)
#    define HAVE_ASYNC_LDS 1
#  endif
#  if __has_builtin(__builtin_amdgcn_s_wait_asynccnt)
#    define HAVE_WAIT_ASYNC 1
#  endif
#endif

// Generic -> addrspace casts via integers:
//  - global VA is numerically identical in AS0/AS1
//  - low 32 bits of a generic LDS pointer are the LDS byte offset (ISA §10.2)
#define AS1V4(p) ((__attribute__((address_space(1))) v4i_vs*)(uintptr_t)(p))
#define AS3V4(p) ((__attribute__((address_space(3))) v4i_vs*)(uint32_t)(uintptr_t)(p))

// ---------------------------------------------------------------------------
// Stage 1: per (b,n) row -> hl, hr (ReLU), then t[k,:] = hl @ W_bil[k]
//   hrW: [B*N, 128]      tW: [B*N, 2, 128]
// ---------------------------------------------------------------------------
__global__ __launch_bounds__(128) void bpl_stage1(
    const float* __restrict__ x,      // [B,N,64]
    const float* __restrict__ W_l,    // [128,192]
    const float* __restrict__ b_l,    // [128]
    const float* __restrict__ W_r,    // [128,192]
    const float* __restrict__ b_r,    // [128]
    const float* __restrict__ W_bil,  // [2,128,128]
    float* __restrict__ hrW,          // [B*N,128]
    float* __restrict__ tW)           // [B*N,2,128]
{
    const int bn  = blockIdx.x;          // 0..8191
    const int b   = bn >> 10;
    const int n   = bn & (NN - 1);
    const int tid = threadIdx.x;         // 0..127, doubles as h / h'

    __shared__ float xwin[3 * F_IN];     // [x(n-1) | x(n) | x(n+1)], zero-padded
    __shared__ float hl_s[HH];

    for (int idx = tid; idx < 3 * F_IN; idx += 128) {
        const int d  = idx / F_IN;       // 0: n-1, 1: n, 2: n+1
        const int f  = idx % F_IN;
        const int nn = n + d - 1;
        float v = 0.f;
        if (nn >= 0 && nn < NN) v = x[(b * NN + nn) * F_IN + f];
        xwin[idx] = v;
    }
    __syncthreads();

    const float* xm1 = xwin;
    const float* xc  = xwin + F_IN;
    const float* xp1 = xwin + 2 * F_IN;

    // hl uses xl = [x(n), x(n-1), x(n+1)]; hr uses xr = [x(n), x(n+1), x(n-1)]
    float al = b_l[tid], ar = b_r[tid];
    const float* wl = W_l + tid * (3 * F_IN);
    const float* wr = W_r + tid * (3 * F_IN);
    #pragma unroll 4
    for (int f = 0; f < F_IN; ++f) {
        al = fmaf(wl[f],            xc[f],  al);
        al = fmaf(wl[F_IN + f],     xm1[f], al);
        al = fmaf(wl[2 * F_IN + f], xp1[f], al);
        ar = fmaf(wr[f],            xc[f],  ar);
        ar = fmaf(wr[F_IN + f],     xp1[f], ar);
        ar = fmaf(wr[2 * F_IN + f], xm1[f], ar);
    }
    const float hl = al > 0.f ? al : 0.f;
    const float hr = ar > 0.f ? ar : 0.f;
    hrW[bn * HH + tid] = hr;
    hl_s[tid] = hl;
    __syncthreads();

    // t[bn,k,h'] = sum_h hl[h] * W_bil[k,h,h']   (h' = tid, coalesced W_bil reads)
    #pragma unroll
    for (int k = 0; k < KOUT; ++k) {
        float acc = 0.f;
        const float* wb = W_bil + (k * HH) * HH + tid;
        #pragma unroll 8
        for (int h = 0; h < HH; ++h)
            acc = fmaf(hl_s[h], wb[h * HH], acc);
        tW[(bn * KOUT + k) * HH + tid] = acc;
    }
}

// ---------------------------------------------------------------------------
// Stage 2: out[b,i,j,k] = dot(t[b,i,k,:], hr[b,j,:]) + b_bil[k]
// Block: 128 threads = 4 waves; tile 32(i) x 32(j) x 2(k).
// Each wave: one 16x16 (i,j) tile, BOTH k accumulators (B-fragment reused).
// LDS fill via GLOBAL_LOAD_ASYNC_TO_LDS_B128 (ASYNCcnt) when available.
// v_wmma_f32_16x16x4_f32, K loop over 128 in steps of 4.
// ---------------------------------------------------------------------------
__global__ __launch_bounds__(128) void bpl_stage2(
    const float* __restrict__ tW,     // [B*N,2,128]
    const float* __restrict__ hrW,    // [B*N,128]
    const float* __restrict__ b_bil,  // [2]
    float* __restrict__ out)          // [B,N,N,2]
{
    __shared__ float ldsT[KOUT][ITILE][LDP];
    __shared__ float ldsR[JTILE][LDP];

    const int j0  = blockIdx.x * JTILE;
    const int i0  = blockIdx.y * ITILE;
    const int b   = blockIdx.z;
    const int tid = threadIdx.x;

    // --- LDS fill: t tile (2x32x128) and hr tile (32x128) ---
    // t rows: global row g = (b*N+i0)*2 + r, r in [0,64): i = i0 + r/2, k = r&1
    const float* srcT = tW + (size_t)(b * NN + i0) * (KOUT * HH);
    const float* srcR = hrW + (size_t)(b * NN + j0) * HH;

#ifdef HAVE_ASYNC_LDS
    // CDNA5 async copy: global -> LDS without touching VGPRs (ASYNCcnt).
    // 16 B per lane per issue; all 128 lanes active, uniform trip counts.
    for (int idx = tid; idx < 64 * (HH / 4); idx += 128) {
        const int r = idx >> 5;          // HH/4 == 32 float4 per row
        const int q = idx & 31;
        __builtin_amdgcn_global_load_async_to_lds_b128(
            AS1V4(srcT + r * HH + q * 4),
            AS3V4(&ldsT[r & 1][r >> 1][q * 4]),
            0, 0);
    }
    for (int idx = tid; idx < JTILE * (HH / 4); idx += 128) {
        const int r = idx >> 5;
        const int q = idx & 31;
        __builtin_amdgcn_global_load_async_to_lds_b128(
            AS1V4(srcR + r * HH + q * 4),
            AS3V4(&ldsR[r][q * 4]),
            0, 0);
    }
#  ifdef HAVE_WAIT_ASYNC
    __builtin_amdgcn_s_wait_asynccnt(0);
#  else
    asm volatile("s_wait_asynccnt 0" ::: "memory");
#  endif
#else
    // Fallback: classic VGPR-mediated fill (float4 loads + ds stores)
    for (int idx = tid; idx < 64 * (HH / 4); idx += 128) {
        const int r = idx >> 5;
        const int q = idx & 31;
        const float4 v = ((const float4*)srcT)[r * (HH / 4) + q];
        *(float4*)&ldsT[r & 1][r >> 1][q * 4] = v;
    }
    for (int idx = tid; idx < JTILE * (HH / 4); idx += 128) {
        const int r = idx >> 5;
        const int q = idx & 31;
        const float4 v = ((const float4*)srcR)[r * (HH / 4) + q];
        *(float4*)&ldsR[r][q * 4] = v;
    }
#endif
    __syncthreads();

    const int wave = tid >> 5;           // 0..3
    const int lane = tid & 31;
    const int half = lane >> 4;          // lanes 16-31 handle upper K pair / M+8
    const int l16  = lane & 15;
    const int iw   = (wave >> 1) * 16;   // i sub-tile
    const int jw   = (wave & 1) * 16;    // j sub-tile

    // Fragment base pointers per ISA layout:
    //   A (16x4 f32): lane -> M = l16, VGPR r -> K = r + 2*half  => float2 at [M][kk+2*half]
    //   B (4x16 f32): lane -> N = l16, VGPR r -> K = r + 2*half  => float2 at [N][kk+2*half]
    const float* arow0 = &ldsT[0][iw + l16][2 * half];
    const float* arow1 = &ldsT[1][iw + l16][2 * half];
    const float* brow  = &ldsR[jw + l16][2 * half];

    v8f acc0 = {};
    v8f acc1 = {};
    #pragma unroll 4
    for (int kk = 0; kk < HH; kk += 4) {
        const v2f a0 = *(const v2f*)(arow0 + kk);
        const v2f a1 = *(const v2f*)(arow1 + kk);
        const v2f bb = *(const v2f*)(brow  + kk);
        acc0 = __builtin_amdgcn_wmma_f32_16x16x4_f32(
                   false, a0, false, bb, (short)0, acc0, false, false);
        acc1 = __builtin_amdgcn_wmma_f32_16x16x4_f32(
                   false, a1, false, bb, (short)0, acc1, false, false);
    }

    // C/D layout: lane -> N = l16, VGPR r -> M = r + 8*half
    const float bb0 = b_bil[0];
    const float bb1 = b_bil[1];
    const int j = j0 + jw + l16;
    #pragma unroll
    for (int r = 0; r < 8; ++r) {
        const int i = i0 + iw + r + 8 * half;
        float2 v;
        v.x = acc0[r] + bb0;
        v.y = acc1[r] + bb1;
        *(float2*)&out[(((size_t)(b * NN + i)) * NN + j) * KOUT] = v;
    }
}

// ---------------------------------------------------------------------------
extern "C" void kernel_launch(void* const* d_in, const int* in_sizes, int n_in,
                              void* d_out, int out_size, void* d_ws, size_t ws_size,
                              hipStream_t stream) {
    const float* x     = (const float*)d_in[0];
    const float* W_l   = (const float*)d_in[1];
    const float* b_l   = (const float*)d_in[2];
    const float* W_r   = (const float*)d_in[3];
    const float* b_r   = (const float*)d_in[4];
    const float* W_bil = (const float*)d_in[5];
    const float* b_bil = (const float*)d_in[6];
    float* out = (float*)d_out;

    // workspace: hr [8*1024*128] (4 MB) then t [8*1024*2*128] (8 MB)
    float* hrW = (float*)d_ws;
    float* tW  = hrW + (size_t)NB * NN * HH;

    bpl_stage1<<<NB * NN, 128, 0, stream>>>(x, W_l, b_l, W_r, b_r, W_bil, hrW, tW);

    dim3 g2(NN / JTILE, NN / ITILE, NB);
    bpl_stage2<<<g2, 128, 0, stream>>>(tW, hrW, b_bil, out);
}